// Model_90056874262792
// MI455X (gfx1250) — compile-verified
//
#include <hip/hip_runtime.h>

// ---------------------------------------------------------------------------
// 4-qubit variational circuit, batched. Each wave32 simulates 16 circuits:
// state held as two 16(amp) x 16(circuit) f32 matrices (re, im) in the WMMA
// C/D register layout (amp -> VGPR index / lane-half, circuit -> lane).
// Per layer:  state <- H_4x4 * diag(phase_l) * state
//   - diag(phase): 48 precomputed complex constants (pre-kernel, weights only)
//   - H (tensor-product Hadamard, symmetric, +-0.25 with 1/4 scale folded):
//     4 accumulating V_WMMA_F32_16X16X4_F32 per matrix, A operands loaded
//     from a precomputed per-lane table (kills per-layer VALU remat).
// C/D -> B operand relayout via v_permlane16_swap_b32 (one VALU op yields the
// B registers for BOTH the low and high K-chunk); ds_bpermute fallback.
// Layer 0 is special-cased (state still real -> no 0*x junk FMACs).
// ---------------------------------------------------------------------------

typedef float v2f __attribute__((ext_vector_type(2)));
typedef float v8f __attribute__((ext_vector_type(8)));

#define WMMA4(A, B, C) \
    __builtin_amdgcn_wmma_f32_16x16x4_f32(false, (A), false, (B), (short)0, (C), false, false)

#if __has_builtin(__builtin_amdgcn_permlane16_swap)
#define HAVE_PLSWAP 1
typedef unsigned int v2u __attribute__((ext_vector_type(2)));
// a.lanes[16:31] <-> b.lanes[0:15]
__device__ __forceinline__ void plswap(float& a, float& b) {
    v2u r = __builtin_amdgcn_permlane16_swap(__float_as_uint(a), __float_as_uint(b),
                                             false, false);
    a = __uint_as_float(r.x);
    b = __uint_as_float(r.y);
}
#else
#define HAVE_PLSWAP 0
#endif

// value held by partner lane (lane ^ 16)
__device__ __forceinline__ float xor16(float v, bool hi) {
#if HAVE_PLSWAP
    float a = v, b = v;
    plswap(a, b);           // a = {v.lo, v.lo}, b = {v.hi, v.hi}
    return hi ? a : b;
#else
    return __shfl_xor(v, 16, 32);
#endif
}

// Build the four K-chunk B operands from C/D-layout state and run the 8
// accumulating f32 WMMAs of one Hadamard layer (both re and im matrices).
__device__ __forceinline__ void hadamard_wmma(float re[8], float im[8],
                                              const v2f Amat[4]) {
    v8f accR = {}, accI = {};
#if HAVE_PLSWAP
    v2f BR[4], BI[4];
    {
        float q0r = re[0], q2r = re[2]; plswap(q0r, q2r);
        BR[0].x = q0r; BR[2].x = q2r;
        float q1r = re[1], q3r = re[3]; plswap(q1r, q3r);
        BR[0].y = q1r; BR[2].y = q3r;
        float q4r = re[4], q6r = re[6]; plswap(q4r, q6r);
        BR[1].x = q4r; BR[3].x = q6r;
        float q5r = re[5], q7r = re[7]; plswap(q5r, q7r);
        BR[1].y = q5r; BR[3].y = q7r;
        float q0i = im[0], q2i = im[2]; plswap(q0i, q2i);
        BI[0].x = q0i; BI[2].x = q2i;
        float q1i = im[1], q3i = im[3]; plswap(q1i, q3i);
        BI[0].y = q1i; BI[2].y = q3i;
        float q4i = im[4], q6i = im[6]; plswap(q4i, q6i);
        BI[1].x = q4i; BI[3].x = q6i;
        float q5i = im[5], q7i = im[7]; plswap(q5i, q7i);
        BI[1].y = q5i; BI[3].y = q7i;
    }
#pragma unroll
    for (int kc = 0; kc < 4; ++kc) {
        accR = WMMA4(Amat[kc], BR[kc], accR);
        accI = WMMA4(Amat[kc], BI[kc], accI);
    }
#else
    const bool hi = (threadIdx.x & 16) != 0;
#pragma unroll
    for (int kc = 0; kc < 4; ++kc) {
        const int e = 4 * (kc & 1);
        v2f Br, Bi;
        float rx2 = __shfl_xor(re[e + 2], 16, 32);
        float rx3 = __shfl_xor(re[e + 3], 16, 32);
        float rx0 = __shfl_xor(re[e + 0], 16, 32);
        float rx1 = __shfl_xor(re[e + 1], 16, 32);
        float ix2 = __shfl_xor(im[e + 2], 16, 32);
        float ix3 = __shfl_xor(im[e + 3], 16, 32);
        float ix0 = __shfl_xor(im[e + 0], 16, 32);
        float ix1 = __shfl_xor(im[e + 1], 16, 32);
        if (kc < 2) {
            Br.x = hi ? rx2 : re[e];     Br.y = hi ? rx3 : re[e + 1];
            Bi.x = hi ? ix2 : im[e];     Bi.y = hi ? ix3 : im[e + 1];
        } else {
            Br.x = hi ? re[e + 2] : rx0; Br.y = hi ? re[e + 3] : rx1;
            Bi.x = hi ? im[e + 2] : ix0; Bi.y = hi ? im[e + 3] : ix1;
        }
        accR = WMMA4(Amat[kc], Br, accR);
        accI = WMMA4(Amat[kc], Bi, accI);
    }
#endif
#pragma unroll
    for (int v = 0; v < 8; ++v) { re[v] = accR[v]; im[v] = accI[v]; }
}

// Pre-kernel:
//  t in [0,48):  ph[l*16+j] = exp(i * A(l,j)),
//                A(l,j) = sum_i q_i(j)*w[4l+i]*(q_{(i+1)%4}(j)-0.5), q_i = bit(3-i)
//  t in [64,96): per-lane WMMA A-operand table, hmat[lane*8 + 2*kc + {0,1}]
__global__ void qsim_tables(const float* __restrict__ w, float2* __restrict__ ph,
                            float* __restrict__ hmat) {
    int t = threadIdx.x;
    if (t < 48) {
        int l = t >> 4, j = t & 15;
        float A = 0.f;
#pragma unroll
        for (int i = 0; i < 4; ++i) {
            int qc = (j >> (3 - i)) & 1;
            int tw = (i + 1) & 3;
            int qt = (j >> (3 - tw)) & 1;
            if (qc) A += w[4 * l + i] * (qt ? 0.5f : -0.5f);
        }
        ph[t] = make_float2(__cosf(A), __sinf(A));
    } else if (t >= 64 && t < 96) {
        int lane = t - 64;
        int hih = (lane >> 4) & 1, m = lane & 15;
#pragma unroll
        for (int kc = 0; kc < 4; ++kc) {
            int k0 = 4 * kc + (hih ? 2 : 0);   // A VGPR0: K=0 (lo) / K=2 (hi)
            int k1 = 4 * kc + (hih ? 3 : 1);   // A VGPR1: K=1 (lo) / K=3 (hi)
            hmat[lane * 8 + 2 * kc]     = (__popc(m & k0) & 1) ? -0.25f : 0.25f;
            hmat[lane * 8 + 2 * kc + 1] = (__popc(m & k1) & 1) ? -0.25f : 0.25f;
        }
    }
}

__global__ __launch_bounds__(128) void qsim_kernel(const float* __restrict__ x,
                                                   const float2* __restrict__ ph,
                                                   const float* __restrict__ hmat,
                                                   const float* __restrict__ lin_w,
                                                   const float* __restrict__ lin_b,
                                                   float* __restrict__ out) {
    const int lane  = threadIdx.x & 31;
    const int waveG = blockIdx.x * (blockDim.x >> 5) + (threadIdx.x >> 5);
    const int col   = (waveG << 4) + (lane & 15);   // circuit for this lane pair
    const bool hi   = (lane & 16) != 0;             // hi half = amp rows 8..15
    const int m     = lane & 15;

    // ---- angles for this circuit (sample s, row r of the 4x4 reshuffle) ----
    const int s = col >> 2, r = col & 3;
    const int base2 = s * 16 + ((r >> 1) << 3) + ((r & 1) << 1);
    float2 xa = *(const float2*)(x + base2);        // k=0,1
    float2 xb = *(const float2*)(x + base2 + 4);    // k=2,3
    float c0 = __cosf(0.5f * xa.x), s0 = __sinf(0.5f * xa.x);
    float c1 = __cosf(0.5f * xa.y), s1 = __sinf(0.5f * xa.y);
    float c2 = __cosf(0.5f * xb.x), s2 = __sinf(0.5f * xb.x);
    float c3 = __cosf(0.5f * xb.y), s3 = __sinf(0.5f * xb.y);

    // ---- initial product state (real), C/D layout: row = v (+8 if hi) ----
    float t0 = hi ? s0 : c0;                        // q0 from lane half
    float u0 = t0 * c1, u1 = t0 * s1;               // q1
    float a0 = u0 * c2, a1 = u0 * s2, a2 = u1 * c2, a3 = u1 * s2;  // q2
    float re[8], im[8];
    re[0] = a0 * c3; re[1] = a0 * s3; re[2] = a1 * c3; re[3] = a1 * s3;
    re[4] = a2 * c3; re[5] = a2 * s3; re[6] = a3 * c3; re[7] = a3 * s3;

    // ---- Hadamard A-operands: precomputed per-lane table (2x b128) ----
    v2f Amat[4];
    {
        float4 h0 = ((const float4*)(hmat + lane * 8))[0];
        float4 h1 = ((const float4*)(hmat + lane * 8))[1];
        Amat[0].x = h0.x; Amat[0].y = h0.y;
        Amat[1].x = h0.z; Amat[1].y = h0.w;
        Amat[2].x = h1.x; Amat[2].y = h1.y;
        Amat[3].x = h1.z; Amat[3].y = h1.w;
    }

    // ---- layer 0: state still real -> phase multiply is 2 muls/row ----
#pragma unroll
    for (int v = 0; v < 8; ++v) {
        float2 p = ph[hi ? v + 8 : v];
        im[v] = p.y * re[v];
        re[v] = p.x * re[v];
    }
    hadamard_wmma(re, im, Amat);

    // ---- layers 1..2: full complex diag phase then H ----
#pragma unroll
    for (int l = 1; l < 3; ++l) {
#pragma unroll
        for (int v = 0; v < 8; ++v) {
            float2 p = ph[l * 16 + (hi ? v + 8 : v)];
            float nr = p.x * re[v] - p.y * im[v];
            float ni = p.x * im[v] + p.y * re[v];
            re[v] = nr; im[v] = ni;
        }
        hadamard_wmma(re, im, Amat);
    }

    // ---- expectations z_w = sum_{q_w=0} p - sum_{q_w=1} p ----
    float S = 0.f, T1 = 0.f, T2 = 0.f, T3 = 0.f;
#pragma unroll
    for (int v = 0; v < 8; ++v) {
        float pv = re[v] * re[v] + im[v] * im[v];
        S  += pv;
        T1 += ((v >> 2) & 1) ? -pv : pv;
        T2 += ((v >> 1) & 1) ? -pv : pv;
        T3 += (v & 1)        ? -pv : pv;
    }
    float oS = xor16(S, hi);
    float z0 = hi ? (oS - S) : (S - oS);            // q0 splits the halves
    float z1 = T1 + xor16(T1, hi);
    float z2 = T2 + xor16(T2, hi);
    float z3 = T3 + xor16(T3, hi);

    // ---- linear (16 -> 2) + softmax; sample = 4 adjacent lanes ----
    float l0 = z0 * lin_w[4 * r]      + z1 * lin_w[4 * r + 1] +
               z2 * lin_w[4 * r + 2]  + z3 * lin_w[4 * r + 3];
    float l1 = z0 * lin_w[16 + 4 * r] + z1 * lin_w[16 + 4 * r + 1] +
               z2 * lin_w[16 + 4 * r + 2] + z3 * lin_w[16 + 4 * r + 3];
    l0 += __shfl_xor(l0, 1, 32); l0 += __shfl_xor(l0, 2, 32);
    l1 += __shfl_xor(l1, 1, 32); l1 += __shfl_xor(l1, 2, 32);

    if (!hi && (m & 3) == 0) {
        l0 += lin_b[0]; l1 += lin_b[1];
        float mx = fmaxf(l0, l1);
        float e0 = __expf(l0 - mx), e1 = __expf(l1 - mx);
        float inv = 1.f / (e0 + e1);
        ((float2*)out)[s] = make_float2(e0 * inv, e1 * inv);
    }
}

extern "C" void kernel_launch(void* const* d_in, const int* in_sizes, int n_in,
                              void* d_out, int out_size, void* d_ws, size_t ws_size,
                              hipStream_t stream) {
    const float* x     = (const float*)d_in[0];   // (bsz, 16)
    const float* w     = (const float*)d_in[1];   // (12,)
    const float* lin_w = (const float*)d_in[2];   // (2, 16)
    const float* lin_b = (const float*)d_in[3];   // (2,)
    float*       out   = (float*)d_out;           // (bsz, 2)
    float2*      ph    = (float2*)d_ws;                           // 48 float2
    float*       hmat  = (float*)((char*)d_ws + 512);             // 256 floats

    qsim_tables<<<1, 96, 0, stream>>>(w, ph, hmat);

    const int bsz    = in_sizes[0] / 16;          // 262144
    const int nCirc  = bsz * 4;                   // 1,048,576 circuits
    const int waves  = nCirc / 16;                // 16 circuits per wave
    const int blocks = waves / 4;                 // 4 waves (128 thr) per block
    qsim_kernel<<<blocks, 128, 0, stream>>>(x, ph, hmat, lin_w, lin_b, out);
}